// TripletLoss_1348619731378
// MI455X (gfx1250) — compile-verified
//
#include <hip/hip_runtime.h>
#include <math.h>

typedef __attribute__((ext_vector_type(2))) float v2f;
typedef __attribute__((ext_vector_type(8))) float v8f;

#define TILE 16
#define DIM 128
#define KSTEPS (DIM / 4)
#define KCHUNK 8
#define WAVES 4
#define MARGIN 0.3f

__device__ __forceinline__ void merge_max(float& v, int& idx, float v2, int i2) {
  if (v2 > v || (v2 == v && i2 < idx)) { v = v2; idx = i2; }
}
__device__ __forceinline__ void merge_min(float& v, int& idx, float v2, int i2) {
  if (v2 < v || (v2 == v && i2 < idx)) { v = v2; idx = i2; }
}

// K0: row-wise squared norms
__global__ void sqnorm_kernel(const float* __restrict__ x, float* __restrict__ sq, int n) {
  int i = blockIdx.x * blockDim.x + threadIdx.x;
  if (i >= n) return;
  const float4* p = (const float4*)(x + (size_t)i * DIM);
  float s = 0.f;
#pragma unroll
  for (int k = 0; k < DIM / 4; ++k) {
    float4 v = p[k];
    s += v.x * v.x + v.y * v.y + v.z * v.z + v.w * v.w;
  }
  sq[i] = s;
}

__global__ void zero_bins_kernel(float* __restrict__ bins, int n) {
  int i = blockIdx.x * blockDim.x + threadIdx.x;
  if (i < n) bins[i] = 0.f;
}

// K1: fused WMMA Gram tile -> distance -> hard pos/neg mining
// grid = n/16 row blocks, 128 threads (4 waves); wave w handles column tiles w, w+4, ...
__global__ void __launch_bounds__(32 * WAVES, 1)
mine_kernel(const float* __restrict__ x,
            const int* __restrict__ targets,
            const float* __restrict__ sq,
            float* __restrict__ ap, float* __restrict__ an,
            int* __restrict__ posp, int* __restrict__ negp,
            int n) {
  __shared__ float s_pv[WAVES][TILE];
  __shared__ int   s_pi[WAVES][TILE];
  __shared__ float s_nv[WAVES][TILE];
  __shared__ int   s_ni[WAVES][TILE];

  const int tid  = threadIdx.x;
  const int lane = tid & 31;
  const int wave = tid >> 5;
  const int hi   = lane >> 4;   // 0 for lanes 0-15, 1 for lanes 16-31
  const int lo   = lane & 15;
  const int rowbase = blockIdx.x * TILE;

  // A fragment: lane holds X[rowbase+lo][4*kk + 2*hi + {0,1}], register resident.
  v2f a[KSTEPS];
  {
    const float* arow = x + (size_t)(rowbase + lo) * DIM + 2 * hi;
#pragma unroll
    for (int kk = 0; kk < KSTEPS; ++kk)
      a[kk] = *(const v2f*)(arow + 4 * kk);
  }

  // Per-slot row metadata: C/D VGPR v maps to row rowbase + v + 8*hi.
  float sqr[8]; int tr[8];
#pragma unroll
  for (int v = 0; v < 8; ++v) {
    int r  = rowbase + v + 8 * hi;
    sqr[v] = sq[r];
    tr[v]  = targets[r];
  }

  float posv[8], negv[8]; int posi[8], negi[8];
#pragma unroll
  for (int v = 0; v < 8; ++v) {
    posv[v] = -INFINITY; posi[v] = 0x7FFFFFFF;
    negv[v] =  INFINITY; negi[v] = 0x7FFFFFFF;
  }

  const int ntiles = n / TILE;
  for (int ct = wave; ct < ntiles; ct += WAVES) {
    const int colbase = ct * TILE;
    const int c = colbase + lo;                       // this lane's column
    const float* bcol = x + (size_t)c * DIM + 2 * hi; // B[k][n] = X[c][k]
    // Two accumulator chains (even/odd k-steps) to break the WMMA RAW chain.
    v8f acc0 = {0.f, 0.f, 0.f, 0.f, 0.f, 0.f, 0.f, 0.f};
    v8f acc1 = {0.f, 0.f, 0.f, 0.f, 0.f, 0.f, 0.f, 0.f};
#pragma unroll
    for (int kc = 0; kc < KSTEPS; kc += KCHUNK) {
      v2f b[KCHUNK];
#pragma unroll
      for (int j = 0; j < KCHUNK; ++j)            // 8 independent loads in flight
        b[j] = *(const v2f*)(bcol + 4 * (kc + j));
#pragma unroll
      for (int j = 0; j < KCHUNK; j += 2) {
        acc0 = __builtin_amdgcn_wmma_f32_16x16x4_f32(false, a[kc + j],     false, b[j],
                                                     (short)0, acc0, false, false);
        acc1 = __builtin_amdgcn_wmma_f32_16x16x4_f32(false, a[kc + j + 1], false, b[j + 1],
                                                     (short)0, acc1, false, false);
      }
    }
    const int   tc  = targets[c];
    const float sqc = sq[c];
#pragma unroll
    for (int v = 0; v < 8; ++v) {
      float dot = acc0[v] + acc1[v];
      float d2 = sqr[v] + sqc - 2.0f * dot;
      float d  = sqrtf(fmaxf(d2, 1e-12f));
      if (tr[v] == tc) {
        if (d > posv[v]) { posv[v] = d; posi[v] = c; }   // columns ascend -> strict > keeps first
      } else {
        if (d < negv[v]) { negv[v] = d; negi[v] = c; }
      }
    }
  }

  // Reduce across the 16 lanes of each half (xor masks stay within halves).
#pragma unroll
  for (int off = 8; off >= 1; off >>= 1) {
#pragma unroll
    for (int v = 0; v < 8; ++v) {
      float pv = __shfl_xor(posv[v], off, 32);
      int   pi = __shfl_xor(posi[v], off, 32);
      float nv = __shfl_xor(negv[v], off, 32);
      int   ni = __shfl_xor(negi[v], off, 32);
      merge_max(posv[v], posi[v], pv, pi);
      merge_min(negv[v], negi[v], nv, ni);
    }
  }

  if (lo == 0) {  // lane 0 carries rows 0..7, lane 16 carries rows 8..15
#pragma unroll
    for (int v = 0; v < 8; ++v) {
      s_pv[wave][8 * hi + v] = posv[v]; s_pi[wave][8 * hi + v] = posi[v];
      s_nv[wave][8 * hi + v] = negv[v]; s_ni[wave][8 * hi + v] = negi[v];
    }
  }
  __syncthreads();
  if (tid < TILE) {
    float bp = -INFINITY, bn = INFINITY; int bpi = 0x7FFFFFFF, bni = 0x7FFFFFFF;
#pragma unroll
    for (int w = 0; w < WAVES; ++w) {
      merge_max(bp, bpi, s_pv[w][tid], s_pi[w][tid]);
      merge_min(bn, bni, s_nv[w][tid], s_ni[w][tid]);
    }
    int row = rowbase + tid;
    ap[row] = bp; an[row] = bn; posp[row] = bpi; negp[row] = bni;
  }
}

// K2: diff = an - ap; per-block loss/prec partials (deterministic two-stage sum)
__global__ void diff_loss_kernel(const float* __restrict__ ap, const float* __restrict__ an,
                                 float* __restrict__ diff,
                                 float* __restrict__ loss_part, float* __restrict__ prec_part,
                                 int n) {
  __shared__ float sl[256], sp[256];
  int tid = threadIdx.x;
  int i = blockIdx.x * 256 + tid;
  float l = 0.f, p = 0.f;
  if (i < n) {
    float a = ap[i], b = an[i];
    diff[i] = b - a;
    l = fmaxf(a - b + MARGIN, 0.f);
    p = (b > a) ? 1.f : 0.f;
  }
  sl[tid] = l; sp[tid] = p;
  __syncthreads();
  for (int s = 128; s > 0; s >>= 1) {
    if (tid < s) { sl[tid] += sl[tid + s]; sp[tid] += sp[tid + s]; }
    __syncthreads();
  }
  if (tid == 0) { loss_part[blockIdx.x] = sl[0]; prec_part[blockIdx.x] = sp[0]; }
}

// K3: exact order statistics via tie-broken rank counting (O(n^2), L2 resident)
__global__ void rank_kernel(const float* __restrict__ diff, float* __restrict__ tvals,
                            int n, int k1, int k2) {
  __shared__ float s[256];
  int i = blockIdx.x * 256 + threadIdx.x;
  float di = (i < n) ? diff[i] : 0.f;
  int cnt = 0;
  for (int base = 0; base < n; base += 256) {
    __syncthreads();
    s[threadIdx.x] = diff[base + threadIdx.x];
    __syncthreads();
#pragma unroll 8
    for (int j = 0; j < 256; ++j) {
      float dj = s[j];
      int jj = base + j;
      cnt += (dj < di || (dj == di && jj < i)) ? 1 : 0;
    }
  }
  if (i < n) {
    if (cnt == k1)          tvals[0] = di;
    else if (cnt == k1 + 1) tvals[1] = di;
    else if (cnt == k2)     tvals[2] = di;
    else if (cnt == k2 + 1) tvals[3] = di;
  }
}

// K4: selection mask + scatter-add histogram (+1.0 atomics are order-exact) + nsel partials
__global__ void select_scatter_kernel(const float* __restrict__ diff,
                                      const float* __restrict__ tvals,
                                      const int* __restrict__ posp, const int* __restrict__ negp,
                                      float* __restrict__ bins, float* __restrict__ nsel_part,
                                      int n, float f1, float f2) {
  __shared__ float ss[256];
  int tid = threadIdx.x;
  int i = blockIdx.x * 256 + tid;
  float t1 = tvals[0] + f1 * (tvals[1] - tvals[0]);
  float t2 = tvals[2] + f2 * (tvals[3] - tvals[2]);
  float sel = 0.f;
  if (i < n) {
    float dv = diff[i];
    sel = (dv > t1 && dv < t2) ? 1.f : 0.f;
    if (sel != 0.f) {
      atomicAdd(&bins[posp[i]], 1.f);
      atomicAdd(&bins[negp[i]], 1.f);
      atomicAdd(&bins[i], 1.f);
    }
  }
  ss[tid] = sel;
  __syncthreads();
  for (int s = 128; s > 0; s >>= 1) {
    if (tid < s) ss[tid] += ss[tid + s];
    __syncthreads();
  }
  if (tid == 0) nsel_part[blockIdx.x] = ss[0];
}

// K5: scalars + normalized replay histogram; sum(bins) == 3*nsel analytically
__global__ void finalize_kernel(const float* __restrict__ bins,
                                const float* __restrict__ loss_part,
                                const float* __restrict__ prec_part,
                                const float* __restrict__ nsel_part,
                                float* __restrict__ out, int n, int nparts) {
  __shared__ float s_inv;
  if (threadIdx.x == 0) {
    float L = 0.f, P = 0.f, S = 0.f;
    for (int i = 0; i < nparts; ++i) { L += loss_part[i]; P += prec_part[i]; S += nsel_part[i]; }
    out[0] = L / (float)n;
    out[1] = P / (float)n;
    s_inv = 1.0f / (3.0f * S);
  }
  __syncthreads();
  float inv = s_inv;
  for (int i = threadIdx.x; i < n; i += blockDim.x) out[2 + i] = bins[i] * inv;
}

extern "C" void kernel_launch(void* const* d_in, const int* in_sizes, int n_in,
                              void* d_out, int out_size, void* d_ws, size_t ws_size,
                              hipStream_t stream) {
  const float* x       = (const float*)d_in[0];
  const int*   targets = (const int*)d_in[1];
  const int n = in_sizes[1];     // 4096 (DIM fixed at 128 per reference)
  float* out = (float*)d_out;

  float* fws  = (float*)d_ws;
  float* sq   = fws;
  float* ap   = fws + (size_t)n;
  float* an   = fws + (size_t)2 * n;
  float* diff = fws + (size_t)3 * n;
  float* bins = fws + (size_t)4 * n;
  int*   posp = (int*)(fws + (size_t)5 * n);
  int*   negp = (int*)(fws + (size_t)6 * n);
  float* loss_part = fws + (size_t)7 * n;
  float* prec_part = loss_part + 64;
  float* nsel_part = prec_part + 64;
  float* tvals     = nsel_part + 64;

  const int nblk = (n + 255) / 256;

  sqnorm_kernel<<<nblk, 256, 0, stream>>>(x, sq, n);
  zero_bins_kernel<<<nblk, 256, 0, stream>>>(bins, n);
  mine_kernel<<<n / TILE, 32 * WAVES, 0, stream>>>(x, targets, sq, ap, an, posp, negp, n);
  diff_loss_kernel<<<nblk, 256, 0, stream>>>(ap, an, diff, loss_part, prec_part, n);

  // jnp.percentile (linear interpolation): q*(n-1) = k + f
  float q1 = 0.45f * (float)(n - 1), q2 = 0.95f * (float)(n - 1);
  int   k1 = (int)q1,                k2 = (int)q2;
  float f1 = q1 - (float)k1,         f2 = q2 - (float)k2;

  rank_kernel<<<nblk, 256, 0, stream>>>(diff, tvals, n, k1, k2);
  select_scatter_kernel<<<nblk, 256, 0, stream>>>(diff, tvals, posp, negp, bins, nsel_part, n, f1, f2);
  finalize_kernel<<<1, 256, 0, stream>>>(bins, loss_part, prec_part, nsel_part, out, n, nblk);
}